// MultiHeadQGATLayer_19018115187415
// MI455X (gfx1250) — compile-verified
//
#include <hip/hip_runtime.h>
#include <cstdint>

typedef __attribute__((ext_vector_type(2))) float v2f;
typedef __attribute__((ext_vector_type(8))) float v8f;

constexpr int IN_DIM   = 256;
constexpr int N_HEADS  = 4;
constexpr int D_HEAD   = 32;
constexpr int OUT_COLS = N_HEADS * D_HEAD;   // 128
constexpr float NEG_SLOPE = 0.01f;

// Order-preserving int encoding of float for atomicMax-based segment max.
__device__ __forceinline__ int enc_f32(float x) {
  int ix = __float_as_int(x);
  return ix >= 0 ? ix : (ix ^ 0x7FFFFFFF);
}
__device__ __forceinline__ float dec_f32(int k) {
  return __int_as_float(k >= 0 ? k : (k ^ 0x7FFFFFFF));
}

// ---------------------------------------------------------------------------
// Init: zero out + denom, set max-keys to INT_MIN.
// ---------------------------------------------------------------------------
__global__ void k_init(float* __restrict__ out, float* __restrict__ denom,
                       int* __restrict__ mkey, int nOut, int nSeg) {
  int i = blockIdx.x * blockDim.x + threadIdx.x;
  int stride = gridDim.x * blockDim.x;
  for (int j = i; j < nOut; j += stride) out[j] = 0.f;
  for (int j = i; j < nSeg; j += stride) { denom[j] = 0.f; mkey[j] = (int)0x80000000; }
}

// ---------------------------------------------------------------------------
// GEMM: z[N x 128] = h[N x 256] * W[256 x 128], W[i][h*32+d] = fc_w[h][i][d].
// One wave per 16x16 tile of z via V_WMMA_F32_16X16X4_F32 (full fp32).
// Block = 256 threads = 8 waves = the 8 column tiles of one 16-row stripe.
// ---------------------------------------------------------------------------
__global__ __launch_bounds__(256) void k_gemm(const float* __restrict__ h,
                                              const float* __restrict__ fc_w,
                                              float* __restrict__ z, int N) {
  constexpr int PAD = 260;               // padded row stride dodges LDS bank conflicts
  __shared__ float sA[16 * PAD];         // 16 rows x 256 cols of h
  const int t = threadIdx.x;
  const int rowBase = blockIdx.x * 16;

  // Cooperative, coalesced float4 load of the 16x256 A tile.
  #pragma unroll
  for (int i = 0; i < 4; ++i) {
    int f4   = t + i * 256;              // 0..1023 float4 slots
    int row  = f4 >> 6;                  // 64 float4 per row
    int col4 = f4 & 63;
    int gRow = rowBase + row;
    if (gRow >= N) gRow = N - 1;         // clamp; tail rows never stored
    float4 v = reinterpret_cast<const float4*>(h + (size_t)gRow * IN_DIM)[col4];
    float* p = &sA[row * PAD + col4 * 4];
    p[0] = v.x; p[1] = v.y; p[2] = v.z; p[3] = v.w;
  }
  __syncthreads();

  const int wave = t >> 5;
  const int lane = t & 31;
  const int half = lane >> 4;            // K sub-block selector per ISA layout
  const int idx  = lane & 15;            // M (for A) / N (for B,C,D)
  const int head = wave >> 1;            // tileN = wave*16 -> head = wave/2
  const int dcol = (wave & 1) * 16 + idx;
  const float* Bb = fc_w + (size_t)head * IN_DIM * D_HEAD + dcol;

  v8f c = {0.f, 0.f, 0.f, 0.f, 0.f, 0.f, 0.f, 0.f};
  const float* aRow = &sA[idx * PAD];

  #pragma unroll 8
  for (int k0 = 0; k0 < IN_DIM; k0 += 4) {
    const int ka = k0 + 2 * half;        // lanes 0-15: K=k0,k0+1; lanes 16-31: K=k0+2,k0+3
    v2f a; a.x = aRow[ka]; a.y = aRow[ka + 1];
    v2f b; b.x = Bb[(size_t)ka * D_HEAD]; b.y = Bb[(size_t)(ka + 1) * D_HEAD];
    c = __builtin_amdgcn_wmma_f32_16x16x4_f32(false, a, false, b,
                                              (short)0, c, false, false);
  }

  // C/D layout: VGPR r holds M=r (lanes 0-15) and M=r+8 (lanes 16-31), N=idx.
  const int colN = wave * 16 + idx;
  #pragma unroll
  for (int r = 0; r < 8; ++r) {
    int row = rowBase + r + 8 * half;
    if (row < N) z[(size_t)row * OUT_COLS + colN] = c[r];
  }
}

// ---------------------------------------------------------------------------
// Per-node attention scalars: s_src[n,h] = z[n,h,:].a_src[h], same for dst.
// Wave per node; lane owns 4 contiguous cols; 8-lane shuffle reduce per head.
// ---------------------------------------------------------------------------
__global__ __launch_bounds__(256) void k_scores(const float* __restrict__ z,
                                                const float* __restrict__ attn_w,
                                                float* __restrict__ ssrc,
                                                float* __restrict__ sdst, int N) {
  const int wave = threadIdx.x >> 5, lane = threadIdx.x & 31;
  const int n = blockIdx.x * 8 + wave;
  if (n >= N) return;                    // uniform across the wave
  const int head = lane >> 3;
  const int dim  = (lane & 7) * 4;
  float4 v = reinterpret_cast<const float4*>(z + (size_t)n * OUT_COLS)[lane];
  const float* as = attn_w + head * 2 * D_HEAD;
  const float* ad = as + D_HEAD;
  float ps = v.x*as[dim] + v.y*as[dim+1] + v.z*as[dim+2] + v.w*as[dim+3];
  float pd = v.x*ad[dim] + v.y*ad[dim+1] + v.z*ad[dim+2] + v.w*ad[dim+3];
  #pragma unroll
  for (int m = 1; m < 8; m <<= 1) {
    ps += __shfl_xor(ps, m, 8);
    pd += __shfl_xor(pd, m, 8);
  }
  if ((lane & 7) == 0) { ssrc[n * N_HEADS + head] = ps; sdst[n * N_HEADS + head] = pd; }
}

// ---------------------------------------------------------------------------
// Edge logits + LeakyReLU + segment max (int-encoded float atomicMax).
// ---------------------------------------------------------------------------
__global__ void k_edge_max(const float* __restrict__ ssrc, const float* __restrict__ sdst,
                           const int* __restrict__ src, const int* __restrict__ dst,
                           float* __restrict__ ebuf, int* __restrict__ mkey, int E) {
  int e = blockIdx.x * blockDim.x + threadIdx.x;
  if (e >= E) return;
  int s = src[e], d = dst[e];
  #pragma unroll
  for (int h2 = 0; h2 < N_HEADS; ++h2) {
    float ev = ssrc[s * N_HEADS + h2] + sdst[d * N_HEADS + h2];
    ev = ev >= 0.f ? ev : NEG_SLOPE * ev;
    ebuf[(size_t)e * N_HEADS + h2] = ev;
    atomicMax(&mkey[d * N_HEADS + h2], enc_f32(ev));
  }
}

// ---------------------------------------------------------------------------
// ex = exp(e - m[dst]); denom[dst] += ex (hardware f32 atomic).
// ---------------------------------------------------------------------------
__global__ void k_edge_exp(const int* __restrict__ dst, const int* __restrict__ mkey,
                           float* __restrict__ ebuf, float* __restrict__ denom, int E) {
  int e = blockIdx.x * blockDim.x + threadIdx.x;
  if (e >= E) return;
  int d = dst[e];
  #pragma unroll
  for (int h2 = 0; h2 < N_HEADS; ++h2) {
    float m  = dec_f32(mkey[d * N_HEADS + h2]);
    float ex = __expf(ebuf[(size_t)e * N_HEADS + h2] - m);
    ebuf[(size_t)e * N_HEADS + h2] = ex;
    unsafeAtomicAdd(&denom[d * N_HEADS + h2], ex);
  }
}

// ---------------------------------------------------------------------------
// out[dst] += alpha * z[src]; wave per edge, lane carries 4 columns.
// ---------------------------------------------------------------------------
__global__ __launch_bounds__(256) void k_aggregate(const float* __restrict__ z,
                                                   const float* __restrict__ ebuf,
                                                   const float* __restrict__ denom,
                                                   const int* __restrict__ src,
                                                   const int* __restrict__ dst,
                                                   float* __restrict__ out, int E) {
  const int wave = threadIdx.x >> 5, lane = threadIdx.x & 31;
  const long e = (long)blockIdx.x * 8 + wave;
  if (e >= E) return;                    // uniform across the wave
  const int s = src[e], d = dst[e];
  const int head = lane >> 3;
  const float alpha = ebuf[e * N_HEADS + head] / denom[d * N_HEADS + head];
  float4 v = reinterpret_cast<const float4*>(z + (size_t)s * OUT_COLS)[lane];
  float* op = out + (size_t)d * OUT_COLS + lane * 4;
  unsafeAtomicAdd(op + 0, alpha * v.x);
  unsafeAtomicAdd(op + 1, alpha * v.y);
  unsafeAtomicAdd(op + 2, alpha * v.z);
  unsafeAtomicAdd(op + 3, alpha * v.w);
}

// ---------------------------------------------------------------------------
extern "C" void kernel_launch(void* const* d_in, const int* in_sizes, int n_in,
                              void* d_out, int out_size, void* d_ws, size_t ws_size,
                              hipStream_t stream) {
  const float* h      = (const float*)d_in[0];
  const float* fc_w   = (const float*)d_in[1];
  const float* attn_w = (const float*)d_in[2];
  const int*   src    = (const int*)d_in[3];
  const int*   dst    = (const int*)d_in[4];
  const int N = in_sizes[0] / IN_DIM;
  const int E = in_sizes[3];
  float* out = (float*)d_out;

  // Workspace layout (fp32 unless noted):
  float* z     = (float*)d_ws;                                // N*128
  float* ssrc  = z     + (size_t)N * OUT_COLS;                // N*4
  float* sdst  = ssrc  + (size_t)N * N_HEADS;                 // N*4
  int*   mkey  = (int*)(sdst + (size_t)N * N_HEADS);          // N*4 (int keys)
  float* denom = (float*)(mkey + (size_t)N * N_HEADS);        // N*4
  float* ebuf  = denom + (size_t)N * N_HEADS;                 // E*4

  k_init     <<<512, 256, 0, stream>>>(out, denom, mkey, N * OUT_COLS, N * N_HEADS);
  k_gemm     <<<(N + 15) / 16, 256, 0, stream>>>(h, fc_w, z, N);
  k_scores   <<<(N + 7) / 8, 256, 0, stream>>>(z, attn_w, ssrc, sdst, N);
  k_edge_max <<<(E + 255) / 256, 256, 0, stream>>>(ssrc, sdst, src, dst, ebuf, mkey, E);
  k_edge_exp <<<(E + 255) / 256, 256, 0, stream>>>(dst, mkey, ebuf, denom, E);
  k_aggregate<<<(E + 7) / 8, 256, 0, stream>>>(z, ebuf, denom, src, dst, out, E);
}